// NODE_32933809225766
// MI455X (gfx1250) — compile-verified
//
#include <hip/hip_runtime.h>
#include <hip/hip_bf16.h>
#include <math.h>

// ---------------------------------------------------------------------------
// Graph GRU-ODE forward on gfx1250 using fp32 WMMA (V_WMMA_F32_16X16X4_F32).
// Sequential scan -> one fused kernel launch per Euler substep / GRU step.
// Each wave owns one 16-col tile and 2-4 row tiles: B fragment loaded once
// per k-step, reused across independent accumulator chains.
// ---------------------------------------------------------------------------

typedef float v2f __attribute__((ext_vector_type(2)));
typedef float v8f __attribute__((ext_vector_type(8)));

#define Bn   8
#define Tn   50
#define Vn   256
#define DIN  32
#define Dd   64
#define Hh   128
#define BLK  64                 // v-rows per workgroup
#define NVB  (Vn / BLK)         // 4
#define WAVES 8
#define NTI  197                // 1 + 49*4 trajectory points
#define DT_  0.25f

// D = A(16x4,f32) * B(4x16,f32) + C(16x16,f32)
__device__ __forceinline__ v8f wmma4(v2f a, v2f b, v8f c) {
  return __builtin_amdgcn_wmma_f32_16x16x4_f32(false, a, false, b, (short)0, c,
                                               false, false);
}

__device__ __forceinline__ float fast_tanh(float x) {
#if __has_builtin(__builtin_amdgcn_tanhf)
  return __builtin_amdgcn_tanhf(x);        // v_tanh_f32 (gfx1250)
#else
  return tanhf(x);
#endif
}
__device__ __forceinline__ float fast_sigmoid(float x) {
  return 1.0f / (1.0f + __expf(-x));       // v_exp_f32 based
}

// ---------------------------------------------------------------------------
// LDS->LDS dense layer: out = tanh(in @ W + bias). Per wave: fixed column
// tile ct, NR row tiles, W fragment shared across the NR accumulator chains.
// ---------------------------------------------------------------------------
template <int Kd, int Nd>
__device__ __forceinline__ void layer_t(const float* bufIn, float* bufOut,
                                        const float* __restrict__ W,
                                        const float* __restrict__ bias,
                                        int wave, int l16, int lh) {
  constexpr int nct   = Nd / 16;
  constexpr int NR    = (4 * nct) / WAVES;   // 4 (Nd=128) or 2 (Nd=64)
  constexpr int rstep = WAVES / nct;         // 1 or 2
  const int ct  = wave % nct;
  const int rt0 = wave / nct;
  const int cc  = ct * 16 + l16;
  const float* Wc = W + cc;

  v8f acc[NR];
#pragma unroll
  for (int i = 0; i < NR; ++i) acc[i] = (v8f){};

#pragma unroll 4
  for (int kc = 0; kc < Kd / 4; ++kc) {
    const int k0 = kc * 4 + 2 * lh;
    v2f bb = { Wc[(size_t)k0 * Nd], Wc[(size_t)(k0 + 1) * Nd] };
#pragma unroll
    for (int i = 0; i < NR; ++i) {
      const float* inRow = bufIn + ((rt0 + rstep * i) * 16 + l16) * Hh;
      v2f a = { inRow[k0], inRow[k0 + 1] };
      acc[i] = wmma4(a, bb, acc[i]);
    }
  }
  const float bv = bias[cc];
#pragma unroll
  for (int i = 0; i < NR; ++i)
#pragma unroll
    for (int j = 0; j < 8; ++j)
      bufOut[((rt0 + rstep * i) * 16 + j + 8 * lh) * Hh + cc] =
          fast_tanh(acc[i][j] + bv);
  __syncthreads();
}

// ---------------------------------------------------------------------------
// Fused Euler substep: h_new = h + DT * tanh-MLP( adj @ h )
// ---------------------------------------------------------------------------
__global__ __launch_bounds__(256) void euler_kernel(
    const float* __restrict__ h_in, int hT, int ht,   // state (B,hT,V,D); null -> h0
    const float* __restrict__ h0,                     // (D)
    const float* __restrict__ adj,                    // (B,V,V)
    const float* __restrict__ Win,  const float* __restrict__ bin,   // (D,H),(H)
    const float* __restrict__ Wh,   const float* __restrict__ bh,    // (2,H,H),(2,H)
    const float* __restrict__ Wout, const float* __restrict__ bout,  // (H,D),(D)
    float* __restrict__ hv_pre, int ti,               // (B,NTI,V,D)
    float* __restrict__ hv_preN, int n)               // (B,Tn,V,D) or null
{
  __shared__ float sm[2 * BLK * Hh];                  // 64 KB ping-pong
  float* bufA = sm;
  float* bufB = sm + BLK * Hh;

  const int b    = blockIdx.x / NVB;
  const int v0   = (blockIdx.x % NVB) * BLK;
  const int wave = threadIdx.x >> 5;
  const int lane = threadIdx.x & 31;
  const int l16  = lane & 15;
  const int lh   = lane >> 4;

  const float* adj_b = adj + (size_t)b * Vn * Vn;
  const float* hb = h_in ? (h_in + (((size_t)b * hT + ht) * Vn) * Dd) : nullptr;

  // ---- stage 1: z0 = adj_blk(64x256) @ h_b(256x64), K=256 -----------------
  // wave -> column tile (wave&3), row tiles {wave>>2, wave>>2 + 2}
  {
    const int ct  = wave & 3;
    const int rt0 = wave >> 2;
    const int cc  = ct * 16 + l16;
    const float* arow0 = adj_b + (size_t)(v0 + rt0 * 16 + l16) * Vn;
    const float* arow1 = adj_b + (size_t)(v0 + (rt0 + 2) * 16 + l16) * Vn;
    v8f acc0 = {}, acc1 = {};

    if (hb) {                                   // branch-free inner loops
      const float* hc = hb + cc;
#pragma unroll 4
      for (int kc = 0; kc < Vn / 4; ++kc) {
        const int k0 = kc * 4 + 2 * lh;
        v2f bb = { hc[(size_t)k0 * Dd], hc[(size_t)(k0 + 1) * Dd] };
        v2f a0 = { arow0[k0], arow0[k0 + 1] };
        v2f a1 = { arow1[k0], arow1[k0 + 1] };
        acc0 = wmma4(a0, bb, acc0);
        acc1 = wmma4(a1, bb, acc1);
      }
    } else {                                    // h0 broadcast: B invariant
      const float hv = h0[cc];
      v2f bb = { hv, hv };
#pragma unroll 4
      for (int kc = 0; kc < Vn / 4; ++kc) {
        const int k0 = kc * 4 + 2 * lh;
        v2f a0 = { arow0[k0], arow0[k0 + 1] };
        v2f a1 = { arow1[k0], arow1[k0 + 1] };
        acc0 = wmma4(a0, bb, acc0);
        acc1 = wmma4(a1, bb, acc1);
      }
    }
#pragma unroll
    for (int j = 0; j < 8; ++j) {
      bufA[(rt0 * 16 + j + 8 * lh) * Hh + cc]       = acc0[j];
      bufA[((rt0 + 2) * 16 + j + 8 * lh) * Hh + cc] = acc1[j];
    }
  }
  __syncthreads();

  layer_t<Dd, Hh>(bufA, bufB, Win,          bin,     wave, l16, lh);
  layer_t<Hh, Hh>(bufB, bufA, Wh,           bh,      wave, l16, lh);
  layer_t<Hh, Hh>(bufA, bufB, Wh + Hh * Hh, bh + Hh, wave, l16, lh);
  layer_t<Hh, Dd>(bufB, bufA, Wout,         bout,    wave, l16, lh);

  // ---- h_new = h + DT*z4 ; write hv_pre[ti] (and hv_preN[n]) --------------
  for (int i = threadIdx.x; i < BLK * Dd; i += 256) {
    const int r = i / Dd, c = i % Dd;
    const int v = v0 + r;
    const float hold = hb ? hb[(size_t)v * Dd + c] : h0[c];
    const float hn = hold + DT_ * bufA[r * Hh + c];
    hv_pre[(((size_t)b * NTI + ti) * Vn + v) * Dd + c] = hn;
    if (hv_preN)
      hv_preN[(((size_t)b * Tn + n) * Vn + v) * Dd + c] = hn;
  }
}

// ---------------------------------------------------------------------------
// GRU correction at observation n: hv_post = hv1*(1-m) + gru(x,hv1)*m
// ---------------------------------------------------------------------------
__global__ __launch_bounds__(256) void gru_kernel(
    const float* __restrict__ h_in, int hT, int ht,   // hv1 = (B,hT,V,D)
    const float* __restrict__ values, const float* __restrict__ masks, // (B,T,V,32)
    const float* __restrict__ adj,
    const float* __restrict__ Wr, const float* __restrict__ br,
    const float* __restrict__ Wz, const float* __restrict__ bz,
    const float* __restrict__ Wg, const float* __restrict__ bg,      // (96,64),(64)
    float* __restrict__ hv_postN, int n)              // (B,Tn,V,D)
{
  __shared__ float xh[BLK * 96];        // [x | h_agg] (stride 96)
  __shared__ float rbuf[BLK * Dd];      // r, later g
  __shared__ float zbuf[BLK * Dd];
  __shared__ float mrow[BLK];

  const int b    = blockIdx.x / NVB;
  const int v0   = (blockIdx.x % NVB) * BLK;
  const int wave = threadIdx.x >> 5;
  const int lane = threadIdx.x & 31;
  const int l16  = lane & 15;
  const int lh   = lane >> 4;
  const int ct   = wave & 3;
  const int rt0  = wave >> 2;
  const int cc   = ct * 16 + l16;

  const float* hb    = h_in   + (((size_t)b * hT + ht) * Vn) * Dd;
  const float* xb    = values + (((size_t)b * Tn + n) * Vn) * DIN;
  const float* mb    = masks  + (((size_t)b * Tn + n) * Vn) * DIN;
  const float* adj_b = adj + (size_t)b * Vn * Vn;

  // X = values*masks into cols [0,32); obs indicator per row
  for (int i = threadIdx.x; i < BLK * DIN; i += 256) {
    const int r = i / DIN, c = i % DIN;
    xh[r * 96 + c] = xb[(size_t)(v0 + r) * DIN + c] * mb[(size_t)(v0 + r) * DIN + c];
  }
  if (threadIdx.x < BLK) {
    float s = 0.f;
    for (int c = 0; c < DIN; ++c) s += fabsf(mb[(size_t)(v0 + threadIdx.x) * DIN + c]);
    mrow[threadIdx.x] = (s > 1e-4f) ? 1.f : 0.f;
  }

  // h_agg = adj_blk @ hv1 into cols [32,96): shared-B, 2 row chains
  {
    const float* arow0 = adj_b + (size_t)(v0 + rt0 * 16 + l16) * Vn;
    const float* arow1 = adj_b + (size_t)(v0 + (rt0 + 2) * 16 + l16) * Vn;
    const float* hc = hb + cc;
    v8f acc0 = {}, acc1 = {};
#pragma unroll 4
    for (int kc = 0; kc < Vn / 4; ++kc) {
      const int k0 = kc * 4 + 2 * lh;
      v2f bb = { hc[(size_t)k0 * Dd], hc[(size_t)(k0 + 1) * Dd] };
      v2f a0 = { arow0[k0], arow0[k0 + 1] };
      v2f a1 = { arow1[k0], arow1[k0 + 1] };
      acc0 = wmma4(a0, bb, acc0);
      acc1 = wmma4(a1, bb, acc1);
    }
#pragma unroll
    for (int j = 0; j < 8; ++j) {
      xh[(rt0 * 16 + j + 8 * lh) * 96 + 32 + cc]       = acc0[j];
      xh[((rt0 + 2) * 16 + j + 8 * lh) * 96 + 32 + cc] = acc1[j];
    }
  }
  __syncthreads();

  // gate = act(xh @ W + b), K=96 (24 k-steps), shared-B, 2 row chains
  auto gate = [&](const float* __restrict__ W, const float* __restrict__ bias,
                  float* outbuf, bool useTanh) {
    const float* Wc = W + cc;
    const float* r0 = xh + (rt0 * 16 + l16) * 96;
    const float* r1 = xh + ((rt0 + 2) * 16 + l16) * 96;
    v8f acc0 = {}, acc1 = {};
#pragma unroll 4
    for (int kc = 0; kc < 24; ++kc) {
      const int k0 = kc * 4 + 2 * lh;
      v2f bb = { Wc[(size_t)k0 * Dd], Wc[(size_t)(k0 + 1) * Dd] };
      v2f a0 = { r0[k0], r0[k0 + 1] };
      v2f a1 = { r1[k0], r1[k0 + 1] };
      acc0 = wmma4(a0, bb, acc0);
      acc1 = wmma4(a1, bb, acc1);
    }
    const float bv = bias[cc];
#pragma unroll
    for (int j = 0; j < 8; ++j) {
      const float x0 = acc0[j] + bv;
      const float x1 = acc1[j] + bv;
      outbuf[(rt0 * 16 + j + 8 * lh) * Dd + cc] =
          useTanh ? fast_tanh(x0) : fast_sigmoid(x0);
      outbuf[((rt0 + 2) * 16 + j + 8 * lh) * Dd + cc] =
          useTanh ? fast_tanh(x1) : fast_sigmoid(x1);
    }
  };

  gate(Wr, br, rbuf, false);
  gate(Wz, bz, zbuf, false);
  __syncthreads();

  // xg: cols [32,96) *= r  (in place)
  for (int i = threadIdx.x; i < BLK * Dd; i += 256) {
    const int r = i / Dd, c = i % Dd;
    xh[r * 96 + 32 + c] *= rbuf[r * Dd + c];
  }
  __syncthreads();

  gate(Wg, bg, rbuf, true);            // g -> rbuf
  __syncthreads();

  // blend and emit
  for (int i = threadIdx.x; i < BLK * Dd; i += 256) {
    const int r = i / Dd, c = i % Dd;
    const int v = v0 + r;
    const float h1 = hb[(size_t)v * Dd + c];
    const float zz = zbuf[r * Dd + c];
    const float gg = rbuf[r * Dd + c];
    const float mm = mrow[r];
    const float gro = (1.f - zz) * h1 + zz * gg;
    hv_postN[(((size_t)b * Tn + n) * Vn + v) * Dd + c] = h1 * (1.f - mm) + gro * mm;
  }
}

// ---------------------------------------------------------------------------
// X_pred[:,1:] = hv_pre_N[:,1:] @ W_out + b_out  (100352x64 @ 64x32), WMMA
// Each wave: one row tile (rt), two column tiles sharing the A fragment.
// ---------------------------------------------------------------------------
#define PROJ_ROWS   (Bn * (Tn - 1) * Vn)              // 100352
#define PROJ_RT     (PROJ_ROWS / 16)                  // 6272 row tiles
#define PROJ_BLOCKS (PROJ_RT / WAVES)                 // 784 (x2 col tiles/wave)
__global__ __launch_bounds__(256) void proj_kernel(
    const float* __restrict__ hv_preN,
    const float* __restrict__ Wout, const float* __restrict__ bout,
    float* __restrict__ xpred)
{
  const int wave = threadIdx.x >> 5;
  const int lane = threadIdx.x & 31;
  const int l16  = lane & 15;
  const int lh   = lane >> 4;
  const int rt   = blockIdx.x * WAVES + wave;         // row tile
  const int ga   = rt * 16 + l16;                     // global A row
  const int bA   = ga / ((Tn - 1) * Vn);
  const int jA   = (ga / Vn) % (Tn - 1);
  const int vA   = ga % Vn;
  const float* arow = hv_preN + (((size_t)bA * Tn + (jA + 1)) * Vn + vA) * Dd;

  v8f acc0 = {}, acc1 = {};
#pragma unroll 4
  for (int kc = 0; kc < Dd / 4; ++kc) {
    const int k0 = kc * 4 + 2 * lh;
    v2f a  = { arow[k0], arow[k0 + 1] };
    v2f b0 = { Wout[(size_t)k0 * DIN + l16],      Wout[(size_t)(k0 + 1) * DIN + l16] };
    v2f b1 = { Wout[(size_t)k0 * DIN + 16 + l16], Wout[(size_t)(k0 + 1) * DIN + 16 + l16] };
    acc0 = wmma4(a, b0, acc0);
    acc1 = wmma4(a, b1, acc1);
  }
  const float bv0 = bout[l16];
  const float bv1 = bout[16 + l16];
#pragma unroll
  for (int j = 0; j < 8; ++j) {
    const size_t g = (size_t)(rt * 16 + j + 8 * lh);
    xpred[g * DIN + l16]      = acc0[j] + bv0;
    xpred[g * DIN + 16 + l16] = acc1[j] + bv1;
  }
}

__global__ void traj_kernel(float* __restrict__ traj) {
  const int i = threadIdx.x;
  if (i < NTI)
    traj[i] = (i == NTI - 1) ? 49.0f : (float)(i >> 2) + 0.25f * (float)(i & 3);
}

// ---------------------------------------------------------------------------
extern "C" void kernel_launch(void* const* d_in, const int* in_sizes, int n_in,
                              void* d_out, int out_size, void* d_ws, size_t ws_size,
                              hipStream_t stream) {
  (void)in_sizes; (void)n_in; (void)out_size; (void)d_ws; (void)ws_size;

  const float* values = (const float*)d_in[0];
  const float* masks  = (const float*)d_in[1];
  const float* adj    = (const float*)d_in[2];
  const float* h0     = (const float*)d_in[3];
  const float* W_out  = (const float*)d_in[4];
  const float* b_out  = (const float*)d_in[5];
  const float* oWin   = (const float*)d_in[6];
  const float* obin   = (const float*)d_in[7];
  const float* oWh    = (const float*)d_in[8];
  const float* obh    = (const float*)d_in[9];
  const float* oWout  = (const float*)d_in[10];
  const float* obout  = (const float*)d_in[11];
  const float* gWr    = (const float*)d_in[12];
  const float* gbr    = (const float*)d_in[13];
  const float* gWz    = (const float*)d_in[14];
  const float* gbz    = (const float*)d_in[15];
  const float* gWg    = (const float*)d_in[16];
  const float* gbg    = (const float*)d_in[17];

  float* out = (float*)d_out;
  const size_t SZ_XPRED = (size_t)Bn * (Tn - 1) * Vn * DIN;   // 3,211,264
  const size_t SZ_HVPRE = (size_t)Bn * NTI * Vn * Dd;         // 25,821,184
  const size_t SZ_HN    = (size_t)Bn * Tn  * Vn * Dd;         // 6,553,600
  float* xpred    = out;
  float* hv_pre   = out + SZ_XPRED;
  float* hv_preN  = hv_pre + SZ_HVPRE;
  float* hv_postN = hv_preN + SZ_HN;
  float* traj     = hv_postN + SZ_HN;

  const dim3 grid(Bn * NVB), blk(256);

  traj_kernel<<<1, 256, 0, stream>>>(traj);

  // t=0: single Euler from broadcast h0, then GRU
  euler_kernel<<<grid, blk, 0, stream>>>(nullptr, 0, 0, h0, adj,
      oWin, obin, oWh, obh, oWout, obout, hv_pre, 0, hv_preN, 0);
  gru_kernel<<<grid, blk, 0, stream>>>(hv_preN, Tn, 0, values, masks, adj,
      gWr, gbr, gWz, gbz, gWg, gbg, hv_postN, 0);

  for (int n = 1; n < Tn; ++n) {
    for (int k = 0; k < 4; ++k) {
      const int ti = 1 + (n - 1) * 4 + k;
      const float* hin; int hT, ht;
      if (k == 0) { hin = hv_postN; hT = Tn;  ht = n - 1; }
      else        { hin = hv_pre;   hT = NTI; ht = ti - 1; }
      euler_kernel<<<grid, blk, 0, stream>>>(hin, hT, ht, h0, adj,
          oWin, obin, oWh, obh, oWout, obout,
          hv_pre, ti, (k == 3 ? hv_preN : nullptr), n);
    }
    gru_kernel<<<grid, blk, 0, stream>>>(hv_preN, Tn, n, values, masks, adj,
        gWr, gbr, gWz, gbz, gWg, gbg, hv_postN, n);
  }

  proj_kernel<<<PROJ_BLOCKS, blk, 0, stream>>>(hv_preN, W_out, b_out, xpred);
}